// DualGraphAttentionNetwork_88948772700936
// MI455X (gfx1250) — compile-verified
//
#include <hip/hip_runtime.h>
#include <hip/hip_bf16.h>

// ---------------------------------------------------------------------------
// CDNA5 (gfx1250) fp32 WMMA building blocks.
// V_WMMA_F32_16X16X4_F32:  D(16x16) = A(16x4) * B(4x16) + C, wave32.
// A frag (v2f): lanes 0-15 -> rows 0-15, K = {0,1};  lanes 16-31 -> K = {2,3}
// B frag (v2f): lanes 0-15 -> cols 0-15, K = {0,1};  lanes 16-31 -> K = {2,3}
// C/D (v8f):    vgpr r, lanes 0-15 -> (row r,   col lane)
//                       lanes 16-31 -> (row r+8, col lane-16)
// ---------------------------------------------------------------------------
typedef __attribute__((ext_vector_type(2))) float v2f;
typedef __attribute__((ext_vector_type(8))) float v8f;

#define NHEAD 4
#define DHEAD 16

// ---------------------------------------------------------------------------
// Generic fp32 WMMA GEMM:  C[M,N] = act(A[M,K] @ W[K,N] + bias) (+ residual)
// Block = 128 threads = 4 waves; wave w computes rows [blk*64 + w*16, +16),
// cols [blockIdx.y*16, +16).
// Fast path (wave-uniform full tiles, K-main multiple of 4): no per-lane
// predication -> straight global_load_b64/b32 + v_wmma.
// Edge tiles / K tails: branchless clamped loads (v_cndmask, no saveexec).
// ---------------------------------------------------------------------------
__global__ void wmma_gemm_f32(const float* __restrict__ A, int lda,
                              const float* __restrict__ W, int ldw,
                              const float* __restrict__ bias,
                              const float* __restrict__ residual, int ldr,
                              float* __restrict__ C, int ldc,
                              int M, int N, int K, int relu)
{
    int wave = threadIdx.x >> 5;
    int lane = threadIdx.x & 31;
    int half = lane >> 4;
    int l    = lane & 15;
    int m0 = (blockIdx.x * 4 + wave) * 16;
    int n0 = blockIdx.y * 16;
    if (m0 >= M) return;                       // wave-uniform exit (EXEC stays full)

    int row = m0 + l;
    int col = n0 + l;
    bool full = (m0 + 16 <= M) && (n0 + 16 <= N);   // wave-uniform
    int Kmain = K & ~3;

    v8f acc = {};
    if (full) {
        const float* aptr = A + (size_t)row * lda + 2 * half;   // 8B-aligned (lda even)
        const float* bptr = W + (size_t)(2 * half) * ldw + col;
        for (int k0 = 0; k0 < Kmain; k0 += 4) {
            v2f a = *(const v2f*)(aptr + k0);                   // global_load_b64
            v2f b;
            b.x = bptr[(size_t)k0 * ldw];
            b.y = bptr[(size_t)(k0 + 1) * ldw];
            acc = __builtin_amdgcn_wmma_f32_16x16x4_f32(false, a, false, b,
                                                        (short)0, acc, false, false);
        }
        if (Kmain < K) {                                        // K tail (<=3 cols)
            int ka  = Kmain + 2 * half;
            int k0c = min(ka, K - 1), k1c = min(ka + 1, K - 1);
            v2f a, b;
            float ax = A[(size_t)row * lda + k0c];
            float ay = A[(size_t)row * lda + k1c];
            float bx = W[(size_t)k0c * ldw + col];
            float by = W[(size_t)k1c * ldw + col];
            a.x = (ka     < K) ? ax : 0.f;
            a.y = (ka + 1 < K) ? ay : 0.f;
            b.x = (ka     < K) ? bx : 0.f;
            b.y = (ka + 1 < K) ? by : 0.f;
            acc = __builtin_amdgcn_wmma_f32_16x16x4_f32(false, a, false, b,
                                                        (short)0, acc, false, false);
        }
        float bv = bias ? bias[col] : 0.f;
        for (int r = 0; r < 8; ++r) {
            int rr = m0 + r + 8 * half;
            float v = acc[r] + bv;
            if (relu) v = fmaxf(v, 0.f);
            if (residual) v += residual[(size_t)rr * ldr + col];
            C[(size_t)rr * ldc + col] = v;
        }
    } else {
        int rc = min(row, M - 1);
        int cc = min(col, N - 1);
        for (int k0 = 0; k0 < K; k0 += 4) {
            int ka  = k0 + 2 * half;
            int k0c = min(ka, K - 1), k1c = min(ka + 1, K - 1);
            v2f a, b;
            float ax = A[(size_t)rc * lda + k0c];
            float ay = A[(size_t)rc * lda + k1c];
            float bx = W[(size_t)k0c * ldw + cc];
            float by = W[(size_t)k1c * ldw + cc];
            a.x = (row < M && ka     < K) ? ax : 0.f;
            a.y = (row < M && ka + 1 < K) ? ay : 0.f;
            b.x = (col < N && ka     < K) ? bx : 0.f;
            b.y = (col < N && ka + 1 < K) ? by : 0.f;
            acc = __builtin_amdgcn_wmma_f32_16x16x4_f32(false, a, false, b,
                                                        (short)0, acc, false, false);
        }
        if (col < N) {
            float bv = bias ? bias[col] : 0.f;
            for (int r = 0; r < 8; ++r) {
                int rr = m0 + r + 8 * half;
                if (rr < M) {
                    float v = acc[r] + bv;
                    if (relu) v = fmaxf(v, 0.f);
                    if (residual) v += residual[(size_t)rr * ldr + col];
                    C[(size_t)rr * ldc + col] = v;
                }
            }
        }
    }
}

// ---------------------------------------------------------------------------
// Fused flash-style attention (one wave per (b, head, 16-query tile)).
// dh == 16 == WMMA tile. Streams 16-key tiles: 4 WMMA for scores, online
// softmax (shfl reductions across 16-lane halves), P re-layout via LDS,
// 4 WMMA for P@V with running rescale. Handles GAT (adj mask + edge bias +
// leaky ReLU) and standard MHA (key-padding mask) via nullable args.
// Q:[B,Nq,64]  K,V:[B,Nk,64]  Out:[B,Nq,64] (+= xp_res if given).
// ---------------------------------------------------------------------------
__global__ void attn_fused(const float* __restrict__ Q,
                           const float* __restrict__ K,
                           const float* __restrict__ V,
                           const float* __restrict__ xp_res,
                           float* __restrict__ Out,
                           const int*   __restrict__ adj,       // [B,Nq,Nk] or null
                           const int*   __restrict__ keypad_z,  // [B,Nk] (mask where ==0) or null
                           const float* __restrict__ eb_elem,   // [B,Nq,Nk,H] or null
                           const float* __restrict__ eb_const,  // [H] or null
                           int Nq, int Nk, int leaky)
{
    int tile = blockIdx.x;
    int ntq  = Nq >> 4;
    int it   = tile % ntq;  tile /= ntq;
    int hh   = tile % NHEAD;
    int b    = tile / NHEAD;
    int lane = threadIdx.x & 31;
    int half = lane >> 4;
    int l    = lane & 15;
    int i0   = it * 16;
    const float scale = 0.25f;                  // 1/sqrt(16)

    __shared__ float plds[16][17];              // padded to dodge bank conflicts

    const float* Qb = Q + (size_t)b * Nq * 64 + hh * DHEAD;
    const float* Kb = K + (size_t)b * Nk * 64 + hh * DHEAD;
    const float* Vb = V + (size_t)b * Nk * 64 + hh * DHEAD;

    v2f aq[4];
    for (int kk = 0; kk < 4; ++kk) {
        int ka = kk * 4 + 2 * half;
        aq[kk].x = Qb[(size_t)(i0 + l) * 64 + ka];
        aq[kk].y = Qb[(size_t)(i0 + l) * 64 + ka + 1];
    }

    float runmax[8], runsum[8];
    for (int r = 0; r < 8; ++r) { runmax[r] = -3.0e38f; runsum[r] = 0.f; }
    v8f cacc = {};

    for (int j0 = 0; j0 < Nk; j0 += 16) {
        // get next K/V tile's lines in flight early (global_prefetch_b8)
        if (j0 + 16 < Nk) {
            __builtin_prefetch(Kb + (size_t)(j0 + 16 + l) * 64, 0, 0);
            __builtin_prefetch(Vb + (size_t)(j0 + 16 + l) * 64, 0, 0);
        }
        // ---- scores S = Q Kt (16x16) ----
        v8f s = {};
        for (int kk = 0; kk < 4; ++kk) {
            int ka = kk * 4 + 2 * half;
            v2f bk;
            bk.x = Kb[(size_t)(j0 + l) * 64 + ka];
            bk.y = Kb[(size_t)(j0 + l) * 64 + ka + 1];
            s = __builtin_amdgcn_wmma_f32_16x16x4_f32(false, aq[kk], false, bk,
                                                      (short)0, s, false, false);
        }
        // ---- bias / leaky / mask / online softmax ----
        int j = j0 + l;
        float corr[8];
        for (int r = 0; r < 8; ++r) {
            int i = i0 + r + 8 * half;
            float sv = s[r] * scale;
            if (eb_elem)       sv += eb_elem[(((size_t)b * Nq + i) * Nk + j) * NHEAD + hh];
            else if (eb_const) sv += eb_const[hh];
            if (leaky) sv = (sv >= 0.f) ? sv : 0.2f * sv;
            bool masked = false;
            if (adj)      masked = (adj[((size_t)b * Nq + i) * Nk + j] == 0);
            if (keypad_z) masked = (keypad_z[(size_t)b * Nk + j] == 0);
            if (masked) sv = -1e9f;

            float rm = sv;
            for (int off = 8; off >= 1; off >>= 1)
                rm = fmaxf(rm, __shfl_xor(rm, off, 16));
            float nm = fmaxf(runmax[r], rm);
            float c  = __expf(runmax[r] - nm);
            float p  = __expf(sv - nm);
            float ps = p;
            for (int off = 8; off >= 1; off >>= 1)
                ps += __shfl_xor(ps, off, 16);
            runsum[r] = runsum[r] * c + ps;
            runmax[r] = nm;
            corr[r]   = c;
            plds[r + 8 * half][l] = p;
        }
        __syncthreads();
        for (int r = 0; r < 8; ++r) cacc[r] *= corr[r];
        // ---- ctx += P @ V_tile ----
        for (int kk = 0; kk < 4; ++kk) {
            int ka = kk * 4 + 2 * half;
            v2f pa, bv;
            pa.x = plds[l][ka];
            pa.y = plds[l][ka + 1];
            bv.x = Vb[(size_t)(j0 + ka)     * 64 + l];
            bv.y = Vb[(size_t)(j0 + ka + 1) * 64 + l];
            cacc = __builtin_amdgcn_wmma_f32_16x16x4_f32(false, pa, false, bv,
                                                         (short)0, cacc, false, false);
        }
        __syncthreads();
    }

    for (int r = 0; r < 8; ++r) {
        int i = i0 + r + 8 * half;
        float v = cacc[r] / runsum[r];
        size_t idx = ((size_t)b * Nq + i) * 64 + hh * DHEAD + l;
        float res = xp_res ? xp_res[idx] : 0.f;
        Out[idx] = res + v;
    }
}

// ---------------------------------------------------------------------------
// LayerNorm over rows (optional elementwise add of a second stream first).
// ---------------------------------------------------------------------------
__global__ void ln_rows_kernel(const float* __restrict__ X,
                               const float* __restrict__ X2, int ldx,
                               float* __restrict__ Y, int ldy,
                               const float* __restrict__ g,
                               const float* __restrict__ bb,
                               int rows, int F)
{
    int r = blockIdx.x * blockDim.x + threadIdx.x;
    if (r >= rows) return;
    const float* x  = X + (size_t)r * ldx;
    const float* x2 = X2 ? X2 + (size_t)r * ldx : nullptr;
    float m = 0.f;
    for (int f = 0; f < F; ++f) m += x[f] + (x2 ? x2[f] : 0.f);
    m /= F;
    float v = 0.f;
    for (int f = 0; f < F; ++f) {
        float d = x[f] + (x2 ? x2[f] : 0.f) - m;
        v += d * d;
    }
    v /= F;
    float inv = rsqrtf(v + 1e-5f);
    float* y = Y + (size_t)r * ldy;
    for (int f = 0; f < F; ++f)
        y[f] = (x[f] + (x2 ? x2[f] : 0.f) - m) * inv * g[f] + bb[f];
}

// Concatenate embedding lookup with dense features: out[r] = [emb[z[r]], dense[r]]
__global__ void build_h0_kernel(const int* __restrict__ z,
                                const float* __restrict__ dense,
                                const float* __restrict__ emb,
                                float* __restrict__ out, int rows, int Fd, int ld)
{
    int r = blockIdx.x * blockDim.x + threadIdx.x;
    if (r >= rows) return;
    int zi = z[r];
    float* o = out + (size_t)r * ld;
    for (int c = 0; c < 16; ++c) o[c] = emb[zi * 16 + c];
    for (int c = 0; c < Fd; ++c) o[16 + c] = dense[(size_t)r * Fd + c];
}

// Drug edge bias: LN(fe=17) -> relu(@We1) -> @We2  per (b,i,j), 4 heads out.
__global__ void edge_bias_drug_kernel(const float* __restrict__ e,
                                      const float* __restrict__ g,
                                      const float* __restrict__ bb,
                                      const float* __restrict__ We1,
                                      const float* __restrict__ be1,
                                      const float* __restrict__ We2,
                                      const float* __restrict__ be2,
                                      float* __restrict__ eb, int total, int fe)
{
    int idx = blockIdx.x * blockDim.x + threadIdx.x;
    if (idx >= total) return;
    const float* x = e + (size_t)idx * fe;
    float m = 0.f; for (int f = 0; f < fe; ++f) m += x[f];
    m /= fe;
    float v = 0.f; for (int f = 0; f < fe; ++f) { float d = x[f] - m; v += d * d; }
    v /= fe;
    float inv = rsqrtf(v + 1e-5f);
    float xn[17];
    for (int f = 0; f < fe; ++f) xn[f] = (x[f] - m) * inv * g[f] + bb[f];
    float hbuf[34];
    int fh = 2 * fe;
    for (int t = 0; t < fh; ++t) {
        float a = be1[t];
        for (int f = 0; f < fe; ++f) a += xn[f] * We1[f * fh + t];
        hbuf[t] = fmaxf(a, 0.f);
    }
    for (int o = 0; o < NHEAD; ++o) {
        float a = be2[o];
        for (int t = 0; t < fh; ++t) a += hbuf[t] * We2[t * NHEAD + o];
        eb[(size_t)idx * NHEAD + o] = a;
    }
}

// Protein edge bias: LN over a single feature collapses to eln_b -> 4 constants.
__global__ void edge_bias_const_kernel(const float* __restrict__ elnb,
                                       const float* __restrict__ We1,
                                       const float* __restrict__ be1,
                                       const float* __restrict__ We2,
                                       const float* __restrict__ be2,
                                       float* __restrict__ ebc)
{
    if (threadIdx.x == 0) {
        float en = elnb[0];
        float h0 = fmaxf(en * We1[0] + be1[0], 0.f);
        float h1 = fmaxf(en * We1[1] + be1[1], 0.f);
        for (int o = 0; o < NHEAD; ++o)
            ebc[o] = be2[o] + h0 * We2[o] + h1 * We2[NHEAD + o];
    }
}

// Masked softmax pooling: out[b,:] = sum_n softmax_n(g)[n] * x[b,n,:]
__global__ void pool_finalize_kernel(const float* __restrict__ x,
                                     const float* __restrict__ g,
                                     const int* __restrict__ z,
                                     float* __restrict__ out, int N)
{
    int b = blockIdx.x;
    int t = threadIdx.x;                // 64 threads
    __shared__ float red[64];
    float mx = -3.0e38f;
    for (int n = t; n < N; n += 64) {
        float gv = (z[(size_t)b * N + n] != 0) ? g[(size_t)b * N + n] : -1e9f;
        mx = fmaxf(mx, gv);
    }
    red[t] = mx; __syncthreads();
    for (int s = 32; s > 0; s >>= 1) {
        if (t < s) red[t] = fmaxf(red[t], red[t + s]);
        __syncthreads();
    }
    mx = red[0]; __syncthreads();
    float sm = 0.f;
    for (int n = t; n < N; n += 64) {
        float gv = (z[(size_t)b * N + n] != 0) ? g[(size_t)b * N + n] : -1e9f;
        sm += __expf(gv - mx);
    }
    red[t] = sm; __syncthreads();
    for (int s = 32; s > 0; s >>= 1) {
        if (t < s) red[t] += red[t + s];
        __syncthreads();
    }
    sm = red[0]; __syncthreads();
    float acc = 0.f;
    for (int n = 0; n < N; ++n) {
        float gv = (z[(size_t)b * N + n] != 0) ? g[(size_t)b * N + n] : -1e9f;
        float a = __expf(gv - mx) / sm;
        acc += a * x[((size_t)b * N + n) * 64 + t];
    }
    out[b * 64 + t] = acc;
}

// Final MLP: [dp|pp](128) -> relu 128 -> relu 64 -> 1, one block per batch.
__global__ void final_mlp_kernel(const float* __restrict__ dp,
                                 const float* __restrict__ pp,
                                 const float* __restrict__ W1, const float* __restrict__ b1,
                                 const float* __restrict__ W2, const float* __restrict__ b2,
                                 const float* __restrict__ W3, const float* __restrict__ b3,
                                 float* __restrict__ out)
{
    int b = blockIdx.x;
    int t = threadIdx.x;                // 128 threads
    __shared__ float c[128], h1[128], h2[64];
    c[t] = (t < 64) ? dp[b * 64 + t] : pp[b * 64 + t - 64];
    __syncthreads();
    float a = b1[t];
    for (int k = 0; k < 128; ++k) a += c[k] * W1[k * 128 + t];
    h1[t] = fmaxf(a, 0.f);
    __syncthreads();
    if (t < 64) {
        float a2 = b2[t];
        for (int k = 0; k < 128; ++k) a2 += h1[k] * W2[k * 64 + t];
        h2[t] = fmaxf(a2, 0.f);
    }
    __syncthreads();
    if (t == 0) {
        float a3 = b3[0];
        for (int k = 0; k < 64; ++k) a3 += h2[k] * W3[k];
        out[b] = a3;
    }
}

// ---------------------------------------------------------------------------
// Host orchestration
// ---------------------------------------------------------------------------
struct GatP {
    const float *We1, *We2, *Wf1, *Wf2, *Wk, *Wn, *Wq, *Wv;
    const float *be1, *be2, *bf1, *bf2, *bn, *elnb, *elng, *ln1b, *ln1g, *ln2b, *ln2g;
};

extern "C" void kernel_launch(void* const* d_in, const int* in_sizes, int n_in,
                              void* d_out, int out_size, void* d_ws, size_t ws_size,
                              hipStream_t stream)
{
    (void)in_sizes; (void)n_in; (void)out_size; (void)ws_size;

    const int B = 8, ND = 64, NP = 1024, FD = 27, FP = 41;
    const int RD = B * ND, RP = B * NP;

    const int*   d_z  = (const int*)d_in[0];
    const float* d_x  = (const float*)d_in[1];
    const float* d_e  = (const float*)d_in[2];
    const int*   d_a  = (const int*)d_in[3];
    const int*   p_z  = (const int*)d_in[4];
    const float* p_xd = (const float*)d_in[5];
    const float* p_e  = (const float*)d_in[6];
    const int*   p_a  = (const int*)d_in[7];
    (void)p_e;   // protein edge LN over 1 feature collapses to eln_b -> constants

    auto F = [&](int i) { return (const float*)d_in[i]; };
    // params pytree (sorted dict keys): cross_d(8..15), cross_p(16..23),
    // drug_emb(24), drug_layers(25+19L), mlp(82..87), ncd_b/g(88,89),
    // ncp_b/g(90,91), pool_d(92..95), pool_p(96..99), prot_emb(100),
    // prot_layers(101+19L).
    auto gat = [&](int base) {
        GatP p;
        p.We1 = F(base + 0);  p.We2 = F(base + 1);  p.Wf1 = F(base + 2);
        p.Wf2 = F(base + 3);  p.Wk  = F(base + 4);  p.Wn  = F(base + 5);
        p.Wq  = F(base + 6);  p.Wv  = F(base + 7);  p.be1 = F(base + 8);
        p.be2 = F(base + 9);  p.bf1 = F(base + 10); p.bf2 = F(base + 11);
        p.bn  = F(base + 12); p.elnb = F(base + 13); p.elng = F(base + 14);
        p.ln1b = F(base + 15); p.ln1g = F(base + 16);
        p.ln2b = F(base + 17); p.ln2g = F(base + 18);
        return p;
    };

    // workspace bump allocator
    float* Wb = (float*)d_ws;
    size_t off = 0;
    auto alloc = [&](size_t n) { float* p = Wb + off; off += n; return p; };
    float* hD  = alloc((size_t)RD * 64);
    float* hP  = alloc((size_t)RP * 64);
    float* xnD = alloc((size_t)RD * 64);
    float* xnP = alloc((size_t)RP * 64);
    float* xpD = alloc((size_t)RD * 64);
    float* xpP = alloc((size_t)RP * 64);
    float* QD  = alloc((size_t)RD * 64);
    float* KD  = alloc((size_t)RD * 64);
    float* VD  = alloc((size_t)RD * 64);
    float* QP  = alloc((size_t)RP * 64);
    float* KP  = alloc((size_t)RP * 64);
    float* VP  = alloc((size_t)RP * 64);
    float* h2D = alloc((size_t)RD * 64);
    float* h2P = alloc((size_t)RP * 64);
    float* f1D = alloc((size_t)RD * 128);
    float* f1P = alloc((size_t)RP * 128);
    float* ebD = alloc((size_t)B * ND * ND * NHEAD);
    float* ebc = alloc(64);
    float* ctxD = alloc((size_t)RD * 64);
    float* ctxP = alloc((size_t)RP * 64);
    float* odB  = alloc((size_t)RD * 64);
    float* opB  = alloc((size_t)RP * 64);
    float* hd2  = alloc((size_t)RD * 64);
    float* hp2  = alloc((size_t)RP * 64);
    float* gD   = alloc((size_t)RD);
    float* gP   = alloc((size_t)RP);
    float* dp   = alloc(B * 64);
    float* pp   = alloc(B * 64);

    auto gemm = [&](const float* A, int lda, const float* Wt, int ldw,
                    const float* bias, const float* res, int ldr,
                    float* C, int ldc, int M, int N, int K, int relu) {
        dim3 g((M + 63) / 64, (N + 15) / 16);
        wmma_gemm_f32<<<g, dim3(128), 0, stream>>>(A, lda, Wt, ldw, bias,
                                                   res, ldr, C, ldc, M, N, K, relu);
    };
    auto ln = [&](const float* X, const float* X2, float* Y,
                  const float* g, const float* bb, int rows, int Ff) {
        ln_rows_kernel<<<(rows + 255) / 256, 256, 0, stream>>>(X, X2, 64, Y, 64,
                                                               g, bb, rows, Ff);
    };

    // ---- build inputs: [emb(z) | dense] ----
    build_h0_kernel<<<(RD + 255) / 256, 256, 0, stream>>>(d_z, d_x, F(24), hD, RD, FD, 64);
    build_h0_kernel<<<(RP + 255) / 256, 256, 0, stream>>>(p_z, p_xd, F(100), hP, RP, FP, 64);

    auto run_gat = [&](float* h, int rows, int Nn, int Fin,
                       const int* adj, const GatP& P,
                       float* xn, float* xp, float* Qb, float* Kb, float* Vb,
                       float* h2, float* f1,
                       const float* ebElem, const float* ebConst) {
        ln(h, nullptr, xn, P.ln1g, P.ln1b, rows, Fin);
        gemm(xn, 64, P.Wn, 64, P.bn, nullptr, 0, xp, 64, rows, 64, Fin, 0);
        gemm(xp, 64, P.Wq, 64, nullptr, nullptr, 0, Qb, 64, rows, 64, 64, 0);
        gemm(xp, 64, P.Wk, 64, nullptr, nullptr, 0, Kb, 64, rows, 64, 64, 0);
        gemm(xp, 64, P.Wv, 64, nullptr, nullptr, 0, Vb, 64, rows, 64, 64, 0);
        int blocks = B * NHEAD * (Nn / 16);
        attn_fused<<<blocks, 32, 0, stream>>>(Qb, Kb, Vb, xp, h2, adj,
                                              (const int*)nullptr,
                                              ebElem, ebConst, Nn, Nn, 1);
        ln(h2, nullptr, xn, P.ln2g, P.ln2b, rows, 64);
        gemm(xn, 64, P.Wf1, 128, P.bf1, nullptr, 0, f1, 128, rows, 128, 64, 1);
        gemm(f1, 128, P.Wf2, 64, P.bf2, h2, 64, h, 64, rows, 64, 128, 0);
    };

    // ---- drug GAT stack ----
    for (int L = 0; L < 3; ++L) {
        GatP P = gat(25 + 19 * L);
        edge_bias_drug_kernel<<<(B * ND * ND + 255) / 256, 256, 0, stream>>>(
            d_e, P.elng, P.elnb, P.We1, P.be1, P.We2, P.be2,
            ebD, B * ND * ND, 17);
        run_gat(hD, RD, ND, (L == 0 ? 16 + FD : 64), d_a, P,
                xnD, xpD, QD, KD, VD, h2D, f1D, ebD, nullptr);
    }
    // ---- protein GAT stack (edge bias collapses to 4 constants) ----
    for (int L = 0; L < 3; ++L) {
        GatP P = gat(101 + 19 * L);
        edge_bias_const_kernel<<<1, 32, 0, stream>>>(P.elnb, P.We1, P.be1,
                                                     P.We2, P.be2, ebc);
        run_gat(hP, RP, NP, (L == 0 ? 16 + FP : 64), p_a, P,
                xnP, xpP, QP, KP, VP, h2P, f1P, nullptr, ebc);
    }

    // ---- cross attention drug<-protein ----
    gemm(hD, 64, F(10), 64, F(14), nullptr, 0, QD, 64, RD, 64, 64, 0);   // Wq,bq
    gemm(hP, 64, F(8),  64, F(12), nullptr, 0, KP, 64, RP, 64, 64, 0);   // Wk,bk
    gemm(hP, 64, F(11), 64, F(15), nullptr, 0, VP, 64, RP, 64, 64, 0);   // Wv,bv
    attn_fused<<<B * NHEAD * (ND / 16), 32, 0, stream>>>(
        QD, KP, VP, nullptr, ctxD, nullptr, p_z, nullptr, nullptr, ND, NP, 0);
    gemm(ctxD, 64, F(9), 64, F(13), nullptr, 0, odB, 64, RD, 64, 64, 0); // Wo,bo
    ln(hD, odB, hd2, F(89), F(88), RD, 64);                              // ncd_g, ncd_b

    // ---- cross attention protein<-drug (uses updated hd2) ----
    gemm(hP, 64,  F(18), 64, F(22), nullptr, 0, QP, 64, RP, 64, 64, 0);
    gemm(hd2, 64, F(16), 64, F(20), nullptr, 0, KD, 64, RD, 64, 64, 0);
    gemm(hd2, 64, F(19), 64, F(23), nullptr, 0, VD, 64, RD, 64, 64, 0);
    attn_fused<<<B * NHEAD * (NP / 16), 32, 0, stream>>>(
        QP, KD, VD, nullptr, ctxP, nullptr, d_z, nullptr, nullptr, NP, ND, 0);
    gemm(ctxP, 64, F(17), 64, F(21), nullptr, 0, opB, 64, RP, 64, 64, 0);
    ln(hP, opB, hp2, F(91), F(90), RP, 64);                              // ncp_g, ncp_b

    // ---- pooling ----
    gemm(hd2, 64, F(92), 128, F(94), nullptr, 0, f1D, 128, RD, 128, 64, 1);
    gemm(f1D, 128, F(93), 1, F(95), nullptr, 0, gD, 1, RD, 1, 128, 0);
    pool_finalize_kernel<<<B, 64, 0, stream>>>(hd2, gD, d_z, dp, ND);
    gemm(hp2, 64, F(96), 128, F(98), nullptr, 0, f1P, 128, RP, 128, 64, 1);
    gemm(f1P, 128, F(97), 1, F(99), nullptr, 0, gP, 1, RP, 1, 128, 0);
    pool_finalize_kernel<<<B, 64, 0, stream>>>(hp2, gP, p_z, pp, NP);

    // ---- final MLP ----
    final_mlp_kernel<<<B, 128, 0, stream>>>(dp, pp, F(82), F(85), F(83), F(86),
                                            F(84), F(87), (float*)d_out);
}